// MHA_78237124264044
// MI455X (gfx1250) — compile-verified
//
#include <hip/hip_runtime.h>

typedef __attribute__((ext_vector_type(2))) float v2f;
typedef __attribute__((ext_vector_type(8))) float v8f;
typedef __attribute__((ext_vector_type(4))) unsigned int u32x4;
typedef __attribute__((ext_vector_type(4))) int i32x4;
typedef __attribute__((ext_vector_type(8))) int i32x8;

#define NB   4
#define TT   2048
#define NH   16
#define HD   64
#define DM   1024            // NH*HD
#define NC   3072            // 3*DM
#define QKV_STRIDE (NB * NH * TT * HD)   // floats per Q/K/V tensor
#define KB   32              // k-block rows staged per TDM transfer
#define NKB  (DM / KB)

// ---------------------------------------------------------------------------
// TDM: DMA a KB x 64 fp32 tile of W (row-major, row stride NC) into LDS.
// D# layout per CDNA5 ISA:
//   group0: [1:0]=count=1, [63:32]=lds_addr, [120:64]=global_addr, [127:126]=type=2
//   group1: [17:16]=data_size=2 (4B), [79:48]=tensor_dim0, [111:80]=tensor_dim1,
//           [127:112]=tile_dim0, [143:128]=tile_dim1, [207:160]=tensor_dim0_stride
// ---------------------------------------------------------------------------
__device__ __forceinline__ void tdm_load_w_tile(const float* gsrc, unsigned int lds_off)
{
    const unsigned long long ga = (unsigned long long)(size_t)gsrc;
    u32x4 g0;
    g0[0] = 1u;                                        // count=1 (valid user D#)
    g0[1] = lds_off;                                   // LDS byte address
    g0[2] = (unsigned int)ga;                          // global_addr[31:0]
    g0[3] = (unsigned int)(ga >> 32) | (2u << 30);     // global_addr[56:32] | type=2
    i32x8 g1;
    g1[0] = 0x20000;                                   // data_size=2 -> 4 bytes
    g1[1] = (int)((NC & 0xFFFF) << 16);                // tensor_dim0[15:0]
    g1[2] = (int)((NC >> 16) | ((DM & 0xFFFF) << 16)); // dim0[31:16] | tensor_dim1[15:0]
    g1[3] = (int)((DM >> 16) | (64 << 16));            // dim1[31:16] | tile_dim0=64
    g1[4] = KB;                                        // tile_dim1=KB (tile_dim2=0)
    g1[5] = NC;                                        // tensor_dim0_stride[31:0]
    g1[6] = 0;                                         // stride hi / dim1_stride lo
    g1[7] = 0;
    const i32x4 z4 = {0, 0, 0, 0};
#if defined(__clang_major__) && __clang_major__ >= 23
    const i32x8 z8 = {0, 0, 0, 0, 0, 0, 0, 0};
    __builtin_amdgcn_tensor_load_to_lds(g0, g1, z4, z4, z8, 0);
#else
    __builtin_amdgcn_tensor_load_to_lds(g0, g1, z4, z4, 0);
#endif
}

// ---------------------------------------------------------------------------
// Kernel 1: QKV projection.  C[8192 x 3072] = X[8192x1024] @ W[1024x3072] + b
// Block = 4 waves sharing one 64-column strip of W (waves take 4 M-tiles).
// W tiles are double-buffered in LDS via the Tensor Data Mover; B-fragments
// come from ds_load, A-fragments stream from global.  Output de-interleaved
// into Q/K/V tensors laid out [N,H,T,D] in d_ws.
// ---------------------------------------------------------------------------
__global__ __launch_bounds__(128)
void qkv_gemm_kernel(const float* __restrict__ X, const float* __restrict__ W,
                     const float* __restrict__ bias, float* __restrict__ qkv)
{
    __shared__ float wtile[2][KB * 64];                // 2 x 8 KB

    const int wid  = threadIdx.x >> 5;
    const int mg   = blockIdx.x / 48;                  // 0..127 (4 M-tiles each)
    const int ng   = blockIdx.x % 48;                  // 0..47  (64-col strip)
    const int row0 = (mg * 4 + wid) * 16;
    const int col0 = ng * 64;
    const int lane = threadIdx.x & 31;
    const int hh   = lane >> 4;
    const int m    = lane & 15;

    // prologue: start DMA of k-block 0
    if (wid == 0)
        tdm_load_w_tile(W + col0, (unsigned int)(size_t)(void*)&wtile[0][0]);

    v8f acc[4] = {};
    const float* xrow = X + (size_t)(row0 + m) * DM + 2 * hh;

    for (int kb = 0; kb < NKB; ++kb) {
        __builtin_amdgcn_s_wait_tensorcnt(0);          // wave0: current tile landed
        __syncthreads();                               // publish tile; drains DScnt
        if (wid == 0 && kb + 1 < NKB)                  // overlap next DMA w/ compute
            tdm_load_w_tile(W + (size_t)(kb + 1) * KB * NC + col0,
                            (unsigned int)(size_t)(void*)&wtile[(kb + 1) & 1][0]);

        const float* wt = &wtile[kb & 1][0];
        const float* xk = xrow + kb * KB;
#pragma unroll
        for (int k = 0; k < KB; k += 4) {
            v2f a;
            { const float2 t = *(const float2*)(xk + k); a[0] = t.x; a[1] = t.y; }
            const float* bp = wt + (k + 2 * hh) * 64 + m;   // rows k+2hh, k+2hh+1
#pragma unroll
            for (int t4 = 0; t4 < 4; ++t4) {
                v2f b;
                b[0] = bp[16 * t4];
                b[1] = bp[64 + 16 * t4];
                acc[t4] = __builtin_amdgcn_wmma_f32_16x16x4_f32(
                    false, a, false, b, (short)0, acc[t4], false, false);
            }
        }
    }

    // De-interleave: flat col c -> (h = c/192, d = (c%192)/3, s = c%3)
#pragma unroll
    for (int t4 = 0; t4 < 4; ++t4) {
        const int c   = col0 + 16 * t4 + m;
        const int h   = c / 192;
        const int rem = c % 192;
        const int d   = rem / 3;
        const int s   = rem % 3;
        const float bv = bias[c];
#pragma unroll
        for (int r = 0; r < 8; ++r) {
            const int row = row0 + 8 * hh + r;
            const int n   = row >> 11;
            const int t   = row & 2047;
            qkv[(size_t)s * QKV_STRIDE +
                (((size_t)(n * NH + h)) * TT + t) * HD + d] = acc[t4][r] + bv;
        }
    }
}

// ---------------------------------------------------------------------------
// Kernel 2: causal flash attention. One wave (= one 32-thread block) handles
// one (n,h) head and one 16-row query tile. fp32 WMMA for S=QK^T and O+=P V,
// online softmax with 16-lane shuffle reductions, P transposed via LDS.
// ---------------------------------------------------------------------------
__global__ __launch_bounds__(32)
void attn_kernel(const float* __restrict__ qkv, float* __restrict__ out)
{
    __shared__ float pbuf[256];          // 16x16 P tile, row-major

    const int qt   = blockIdx.x & 127;   // query tile index
    const int nh   = blockIdx.x >> 7;    // n*16 + h
    const int lane = threadIdx.x & 31;
    const int hh   = lane >> 4;
    const int m    = lane & 15;
    const int q0   = qt * 16;

    const float* Q = qkv + (size_t)nh * TT * HD;
    const float* K = Q + QKV_STRIDE;
    const float* V = Q + 2 * (size_t)QKV_STRIDE;

    // Preload Q A-fragments for all 16 k-steps (Q tile is 16x64 fp32)
    v2f qa[16];
    {
        const float* qrow = Q + (size_t)(q0 + m) * HD + 2 * hh;
#pragma unroll
        for (int kk = 0; kk < 16; ++kk) {
            const float2 t = *(const float2*)(qrow + 4 * kk);
            qa[kk][0] = t.x; qa[kk][1] = t.y;
        }
    }

    v8f o[4] = {};
    float mrow[8], lrow[8];
#pragma unroll
    for (int r = 0; r < 8; ++r) { mrow[r] = -__builtin_inff(); lrow[r] = 0.0f; }

    for (int j = 0; j <= qt; ++j) {
        // ---- S = Q @ K^T  (16 fp32 WMMA k-steps over D=64) ----
        v8f s = {};
        const float* krow = K + (size_t)(j * 16 + m) * HD + 2 * hh;
        if (j < qt) {                                  // prefetch next K tile
            __builtin_prefetch(krow + 16 * HD, 0, 1);
            __builtin_prefetch(V + (size_t)((j + 1) * 16 + m) * HD, 0, 1);
        }
#pragma unroll
        for (int kk = 0; kk < 16; ++kk) {
            const float2 t = *(const float2*)(krow + 4 * kk);
            v2f b; b[0] = t.x; b[1] = t.y;
            s = __builtin_amdgcn_wmma_f32_16x16x4_f32(
                false, qa[kk], false, b, (short)0, s, false, false);
        }

        // ---- scale + causal mask (diagonal tile only) ----
        const bool diag = (j == qt);
        float p[8];
#pragma unroll
        for (int r = 0; r < 8; ++r) {
            float sc = s[r] * 0.125f;                       // 1/sqrt(64)
            if (diag && (16 * j + m > q0 + 8 * hh + r)) sc = -1e9f;
            p[r] = sc;
        }

        // ---- online softmax: row max / exp / row sum over the 16 lanes ----
#pragma unroll
        for (int r = 0; r < 8; ++r) {
            float v = p[r];
            v = fmaxf(v, __shfl_xor(v, 1, 32));
            v = fmaxf(v, __shfl_xor(v, 2, 32));
            v = fmaxf(v, __shfl_xor(v, 4, 32));
            v = fmaxf(v, __shfl_xor(v, 8, 32));
            const float mn    = fmaxf(mrow[r], v);
            const float alpha = expf(mrow[r] - mn);          // 0 on first tile
            mrow[r] = mn;
            p[r] = expf(p[r] - mn);                          // masked -> 0
            float sum = p[r];
            sum += __shfl_xor(sum, 1, 32);
            sum += __shfl_xor(sum, 2, 32);
            sum += __shfl_xor(sum, 4, 32);
            sum += __shfl_xor(sum, 8, 32);
            lrow[r] = lrow[r] * alpha + sum;
            o[0][r] *= alpha; o[1][r] *= alpha;
            o[2][r] *= alpha; o[3][r] *= alpha;
        }

        // ---- P: C/D layout -> A-frag layout via LDS transpose bounce ----
        __syncthreads();                 // single-wave block: cheap ordering
#pragma unroll
        for (int r = 0; r < 8; ++r) pbuf[(8 * hh + r) * 16 + m] = p[r];
        __syncthreads();

        // ---- O += P @ V_tile  (K-depth 16 -> 4 WMMA steps x 4 N-subtiles) ----
        const float* vbase = V + (size_t)(j * 16) * HD;
#pragma unroll
        for (int kk = 0; kk < 4; ++kk) {
            v2f pa;
            {
                const float2 t = *(const float2*)(&pbuf[m * 16 + 4 * kk + 2 * hh]);
                pa[0] = t.x; pa[1] = t.y;
            }
#pragma unroll
            for (int t4 = 0; t4 < 4; ++t4) {
                v2f b;
                b[0] = vbase[(4 * kk + 2 * hh    ) * HD + 16 * t4 + m];
                b[1] = vbase[(4 * kk + 2 * hh + 1) * HD + 16 * t4 + m];
                o[t4] = __builtin_amdgcn_wmma_f32_16x16x4_f32(
                    false, pa, false, b, (short)0, o[t4], false, false);
            }
        }
    }

    // ---- normalize and store: out[n][t][h*64 + d] ----
    const int n = nh >> 4, h = nh & 15;
#pragma unroll
    for (int t4 = 0; t4 < 4; ++t4) {
#pragma unroll
        for (int r = 0; r < 8; ++r) {
            const int trow = q0 + 8 * hh + r;
            out[((size_t)(n * TT + trow)) * DM + h * HD + 16 * t4 + m] =
                o[t4][r] / lrow[r];
        }
    }
}

extern "C" void kernel_launch(void* const* d_in, const int* in_sizes, int n_in,
                              void* d_out, int out_size, void* d_ws, size_t ws_size,
                              hipStream_t stream)
{
    const float* x    = (const float*)d_in[0];   // [N,T,DM] fp32
    // d_in[1] = mask (causality is hardcoded; unused)
    const float* Wq   = (const float*)d_in[2];   // [DM, 3*DM]
    const float* bq   = (const float*)d_in[3];   // [3*DM]
    float*       out  = (float*)d_out;           // [N,T,DM]
    float*       qkvw = (float*)d_ws;            // 3 * QKV_STRIDE floats (96 MB)

    // GEMM: 128 M-groups x 48 N-strips, 4 waves/block sharing the W strip
    qkv_gemm_kernel<<<128 * 48, 128, 0, stream>>>(x, Wq, bq, qkvw);

    // Attention: 64 heads x 128 query tiles, one wave per block
    attn_kernel<<<64 * 128, 32, 0, stream>>>(qkvw, out);
}